// OTStyleModel_8048768713296
// MI455X (gfx1250) — compile-verified
//
#include <hip/hip_runtime.h>

#define B_     128
#define L_     256
#define HID_   768
#define HID2_  1536
#define CDIM_  192
#define SDIM_  128
#define NPROJ_ 128
#define FDIM_  321

#define KCHUNK 32            // K elements staged per TDM transfer
#define NCHUNK (HID_ / KCHUNK)
#define BROW   33            // padded LDS row stride (floats): 32 data + 1 pad

typedef __attribute__((ext_vector_type(2))) float    v2f;
typedef __attribute__((ext_vector_type(8))) float    v8f;
typedef __attribute__((ext_vector_type(4))) unsigned u32x4;
typedef __attribute__((ext_vector_type(8))) int      i32x8;
typedef __attribute__((ext_vector_type(4))) int      i32x4;

// ---- workspace layout (float units) ----
#define OFF_REP   0
#define OFF_FUSED (OFF_REP + B_*HID2_)
#define OFF_P     (OFF_FUSED + B_*320)
#define OFF_CLS   (OFF_P + (size_t)B_*L_*NPROJ_)
#define OFF_CNT   (OFF_CLS + B_*L_)
#define OFF_DOT   (OFF_CNT + B_*4)

// ---------------------------------------------------------------------------
// Kernel A: per-batch class labels, counts, cls row, masked mean diff -> rep
// ---------------------------------------------------------------------------
__global__ void __launch_bounds__(256) k_stats(const float* __restrict__ H,
                                               const int* __restrict__ tt,
                                               const int* __restrict__ m,
                                               float* __restrict__ rep,
                                               int* __restrict__ cls01,
                                               int* __restrict__ counts,
                                               unsigned* __restrict__ d_ot) {
    int b = blockIdx.x;
    int t = threadIdx.x;               // 256 threads, one per l
    __shared__ int cls[L_];
    __shared__ int cnt[2];
    if (t < 2) cnt[t] = 0;
    __syncthreads();

    int mv = m[b*L_ + t];
    int tv = tt[b*L_ + t];
    int c = (mv == 1) ? ((tv == 0) ? 0 : 1) : 2;
    cls[t] = c;
    cls01[b*L_ + t] = c;
    if (c < 2) atomicAdd(&cnt[c], 1);
    __syncthreads();

    float inv0 = 1.0f / fmaxf((float)cnt[0], 1.0f);
    float inv1 = 1.0f / fmaxf((float)cnt[1], 1.0f);

    const float* Hb = H + (size_t)b * L_ * HID_;
    #pragma unroll
    for (int hh = 0; hh < 3; ++hh) {
        int h = t + hh * 256;
        float s0 = 0.f, s1 = 0.f;
        for (int l = 0; l < L_; ++l) {
            float v = Hb[(size_t)l*HID_ + h];   // coalesced over threads
            int cl = cls[l];
            s0 += (cl == 0) ? v : 0.f;
            s1 += (cl == 1) ? v : 0.f;
        }
        rep[(size_t)b*HID2_ + h]         = Hb[h];          // cls token row
        rep[(size_t)b*HID2_ + HID_ + h]  = s0*inv0 - s1*inv1;
    }
    if (t == 0) {
        counts[b*4+0] = cnt[0];
        counts[b*4+1] = cnt[1];
        counts[b*4+2] = (cnt[0] < cnt[1]) ? cnt[0] : cnt[1];
        d_ot[b] = 0u;                                     // for atomicMax
    }
}

// ---------------------------------------------------------------------------
// Kernel B: C = rep@Wc.T + bc, S = rep@Ws.T + bs, gate, fused (320 per batch)
// ---------------------------------------------------------------------------
__global__ void __launch_bounds__(320) k_fused(const float* __restrict__ rep,
                                               const float* __restrict__ Wc,
                                               const float* __restrict__ bc,
                                               const float* __restrict__ Ws,
                                               const float* __restrict__ bs,
                                               const float* __restrict__ gate,
                                               float* __restrict__ fused) {
    int b = blockIdx.x;
    int j = threadIdx.x;               // 0..319
    float g = 1.0f / (1.0f + expf(-gate[0]));
    const float* r = rep + (size_t)b * HID2_;
    const float* w;
    float bias, scale;
    if (j < CDIM_) { w = Wc + (size_t)j*HID2_;          bias = bc[j];        scale = 1.0f - g; }
    else           { w = Ws + (size_t)(j-CDIM_)*HID2_;  bias = bs[j-CDIM_];  scale = g; }
    float acc = 0.f;
    for (int k = 0; k < HID2_; ++k) acc = fmaf(r[k], w[k], acc);
    fused[b*320 + j] = (acc + bias) * scale;
}

// ---------------------------------------------------------------------------
// TDM: stage a KCHUNK x NPROJ panel of proj_dirs into LDS.
// D# per ISA 8.3/8.4 (2-D tensor, groups 2/3 zero):
//  group0: count=1 | lds_addr | global_addr(57b) | type=2
//  group1: data_size=4B, pad_enable=1, pad_interval=4 (32 DWORDs = one row),
//          pad_amount=0 (1 DWORD)  -> LDS row stride 33 floats (bank-friendly)
//          tensor_dim0=HID, tensor_dim1=NPROJ, tile_dim0=KCHUNK,
//          tile_dim1=NPROJ, tensor_dim0_stride=HID
// This toolchain exposes the 6-arg builtin: (g0, g1, g2, g3, g4, cpol).
// ---------------------------------------------------------------------------
__device__ __forceinline__ void tdm_load_b_chunk(const float* pd, int kbase,
                                                 unsigned lds_byte_off) {
    unsigned long long ga = (unsigned long long)(const void*)(pd + kbase);
    u32x4 g0;
    g0[0] = 1u;                                            // count=1 (valid)
    g0[1] = lds_byte_off;                                  // lds_addr
    g0[2] = (unsigned)(ga & 0xFFFFFFFFu);                  // global_addr[31:0]
    g0[3] = (unsigned)((ga >> 32) & 0x1FFFFFFu) | (2u << 30); // [56:32] | type=2
    i32x8 g1;
    g1[0] = (2 << 16) | (1 << 20) | (4 << 22);  // data_size=4B, pad_en, interval=32dw
    g1[1] = (int)((unsigned)(HID_ & 0xFFFF) << 16);        // tensor_dim0[15:0] @63:48
    g1[2] = (int)((unsigned)(NPROJ_ & 0xFFFF) << 16);      // tensor_dim1[15:0] @95:80
    g1[3] = (int)((unsigned)(KCHUNK & 0xFFFF) << 16);      // tile_dim0 @127:112
    g1[4] = NPROJ_;                                        // tile_dim1 @143:128
    g1[5] = HID_;                                          // tensor_dim0_stride lo32
    g1[6] = 0;
    g1[7] = 0;
    i32x4 z4 = {0, 0, 0, 0};
    i32x8 z8 = {0, 0, 0, 0, 0, 0, 0, 0};
    __builtin_amdgcn_tensor_load_to_lds(g0, g1, z4, z4, z8, 0);
}

// ---------------------------------------------------------------------------
// Kernel C: P[b,l,p] = H[b,l,:] . proj_dirs[p,:]  via V_WMMA_F32_16X16X4_F32
// grid (B, 4): block = 64 rows x 128 proj, 8 waves.
// B panel staged to LDS by the Tensor Data Mover, double-buffered; wave 0
// drives the TDM (EXEC-independent, one issue per wave), barriers publish.
// A fragment (16x4 f32): lane 0-15 -> M=lane, K0/K1 ; lane 16-31 -> M, K2/K3
// B fragment (4x16 f32): lane 0-15 -> N=lane, K0/K1 ; lane 16-31 -> N, K2/K3
// ---------------------------------------------------------------------------
__global__ void __launch_bounds__(256) k_proj(const float* __restrict__ H,
                                              const float* __restrict__ pd,
                                              float* __restrict__ P) {
    __shared__ float ldsB[2][NPROJ_ * BROW];               // 2 x 16.5 KB

    int b      = blockIdx.x;
    int lchunk = blockIdx.y;                 // 0..3
    int w      = threadIdx.x >> 5;
    int lane   = threadIdx.x & 31;
    int rt     = w & 3;                      // row tile
    int ch     = w >> 2;                     // col half
    int mn     = lane & 15;                  // M for A, N for B
    int koff   = (lane >> 4) * 2;            // 0 or 2

    const float* Arow = H + (size_t)(b*L_ + lchunk*64 + rt*16 + mn) * HID_;

    unsigned buf_off[2];
    buf_off[0] = (unsigned)(size_t)(void*)&ldsB[0][0];
    buf_off[1] = (unsigned)(size_t)(void*)&ldsB[1][0];

    if (w == 0) tdm_load_b_chunk(pd, 0, buf_off[0]);       // prime chunk 0

    v8f acc[4] = {};
    for (int c = 0; c < NCHUNK; ++c) {
        if (w == 0) {
            if (c + 1 < NCHUNK) {
                tdm_load_b_chunk(pd, (c + 1) * KCHUNK, buf_off[(c + 1) & 1]);
                __builtin_amdgcn_s_wait_tensorcnt(1);      // chunk c done (in-order)
            } else {
                __builtin_amdgcn_s_wait_tensorcnt(0);
            }
        }
        __syncthreads();                                   // publish buf[c&1]

        const float* Bbuf = &ldsB[c & 1][0];
        const float* Abase = Arow + c * KCHUNK;
        #pragma unroll
        for (int kl = 0; kl < KCHUNK; kl += 4) {
            v2f a = *(const v2f*)(Abase + kl + koff);
            #pragma unroll
            for (int t = 0; t < 4; ++t) {
                const float* bp = Bbuf + (ch*64 + t*16 + mn) * BROW + kl + koff;
                v2f bm = *(const v2f*)bp;
                acc[t] = __builtin_amdgcn_wmma_f32_16x16x4_f32(
                    false, a, false, bm, (short)0, acc[t], false, false);
            }
        }
        __syncthreads();                                   // buffer reuse fence
    }

    // C/D layout: VGPR r, lanes 0-15: (M=r, N=lane); lanes 16-31: (M=r+8, N=lane-16)
    float* Pb   = P + (size_t)(b * L_) * NPROJ_;
    int rbase   = lchunk*64 + rt*16 + ((lane >> 4) << 3);
    int cbase   = ch*64 + (lane & 15);
    #pragma unroll
    for (int t = 0; t < 4; ++t) {
        #pragma unroll
        for (int r = 0; r < 8; ++r)
            Pb[(size_t)(rbase + r)*NPROJ_ + t*16 + cbase] = acc[t][r];
    }
}

// ---------------------------------------------------------------------------
// Kernel D: truncated sorted-pair L1 distance, rank-based (no explicit sort).
// grid (B, 16), 8 waves; each wave owns one projection p.
// rank(e) = #{same-class elems strictly smaller (index tie-break)} -> scatter.
// ---------------------------------------------------------------------------
__global__ void __launch_bounds__(256) k_ot(const float* __restrict__ P,
                                            const int* __restrict__ cls01,
                                            const int* __restrict__ counts,
                                            unsigned* __restrict__ d_ot) {
    int b    = blockIdx.x;
    int grp  = blockIdx.y;                   // 0..15
    int w    = threadIdx.x >> 5;
    int lane = threadIdx.x & 31;
    int p    = grp*8 + w;

    __shared__ float         keys[8][L_];
    __shared__ unsigned char sets[8][L_];
    __shared__ float         srt [8][2][L_];

    const float* Pb = P + (size_t)(b*L_)*NPROJ_ + p;
    const int*   cb = cls01 + b*L_;

    float v[8]; int s[8];
    #pragma unroll
    for (int j = 0; j < 8; ++j) {
        int l = j*32 + lane;
        v[j] = Pb[(size_t)l * NPROJ_];
        s[j] = cb[l];
        keys[w][l] = v[j];
        sets[w][l] = (unsigned char)s[j];
    }
    __syncthreads();

    int rank[8] = {0,0,0,0,0,0,0,0};
    for (int lp = 0; lp < L_; ++lp) {
        float vb = keys[w][lp];              // broadcast read, no bank conflict
        int   sb = sets[w][lp];
        #pragma unroll
        for (int j = 0; j < 8; ++j) {
            int l = j*32 + lane;
            bool less = (vb < v[j]) || (vb == v[j] && lp < l);
            rank[j] += (sb == s[j] && less) ? 1 : 0;
        }
    }
    #pragma unroll
    for (int j = 0; j < 8; ++j)
        if (s[j] < 2) srt[w][s[j]][rank[j]] = v[j];
    __syncthreads();

    int mcnt = counts[b*4+2];
    float sum = 0.f;
    for (int i = lane; i < mcnt; i += 32)
        sum += fabsf(srt[w][0][i] - srt[w][1][i]);
    #pragma unroll
    for (int off = 16; off > 0; off >>= 1)
        sum += __shfl_xor(sum, off, 32);
    if (lane == 0) {
        float mean = sum / (float)((mcnt > 1) ? mcnt : 1);
        atomicMax(&d_ot[b], __float_as_uint(mean));   // valid: mean >= 0
    }
}

// ---------------------------------------------------------------------------
// Kernel E: feat=[fused, d_ot], layernorm, classifier. 1 block, thread = batch.
// ---------------------------------------------------------------------------
__global__ void __launch_bounds__(128) k_head(const float* __restrict__ fused,
                                              const unsigned* __restrict__ d_ot,
                                              const float* __restrict__ ln_g,
                                              const float* __restrict__ ln_b,
                                              const float* __restrict__ Wcls,
                                              const float* __restrict__ bcls,
                                              float* __restrict__ out) {
    int b = threadIdx.x;
    const float* f = fused + b*320;
    float last = __uint_as_float(d_ot[b]);

    float sum = last, sq = last*last;
    for (int j = 0; j < 320; ++j) { float x = f[j]; sum += x; sq = fmaf(x, x, sq); }
    float mean = sum / (float)FDIM_;
    float var  = sq / (float)FDIM_ - mean*mean;
    float rs   = rsqrtf(var + 1e-5f);

    float o0 = bcls[0], o1 = bcls[1];
    for (int j = 0; j < FDIM_; ++j) {
        float x  = (j < 320) ? f[j] : last;
        float nv = (x - mean) * rs * ln_g[j] + ln_b[j];
        o0 = fmaf(nv, Wcls[j], o0);
        o1 = fmaf(nv, Wcls[FDIM_ + j], o1);
    }
    out[b*2+0] = o0;
    out[b*2+1] = o1;
}

// ---------------------------------------------------------------------------
extern "C" void kernel_launch(void* const* d_in, const int* in_sizes, int n_in,
                              void* d_out, int out_size, void* d_ws, size_t ws_size,
                              hipStream_t stream) {
    (void)in_sizes; (void)n_in; (void)out_size; (void)ws_size;

    const float* H     = (const float*)d_in[0];
    const int*   tt    = (const int*)  d_in[1];
    const int*   m     = (const int*)  d_in[2];
    const float* Wc    = (const float*)d_in[3];
    const float* bc    = (const float*)d_in[4];
    const float* Ws    = (const float*)d_in[5];
    const float* bs    = (const float*)d_in[6];
    const float* gate  = (const float*)d_in[7];
    const float* ln_g  = (const float*)d_in[8];
    const float* ln_b  = (const float*)d_in[9];
    const float* Wcls  = (const float*)d_in[10];
    const float* bcls  = (const float*)d_in[11];
    const float* pd    = (const float*)d_in[12];

    float*    ws     = (float*)d_ws;
    float*    rep    = ws + OFF_REP;
    float*    fused  = ws + OFF_FUSED;
    float*    P      = ws + OFF_P;
    int*      cls01  = (int*)(ws + OFF_CLS);
    int*      counts = (int*)(ws + OFF_CNT);
    unsigned* d_ot   = (unsigned*)(ws + OFF_DOT);

    k_stats<<<B_, 256, 0, stream>>>(H, tt, m, rep, cls01, counts, d_ot);
    k_fused<<<B_, 320, 0, stream>>>(rep, Wc, bc, Ws, bs, gate, fused);
    k_proj <<<dim3(B_, 4), 256, 0, stream>>>(H, pd, P);
    k_ot   <<<dim3(B_, 16), 256, 0, stream>>>(P, cls01, counts, d_ot);
    k_head <<<1, 128, 0, stream>>>(fused, d_ot, ln_g, ln_b, Wcls, bcls, (float*)d_out);
}